// Critic_59777354826141
// MI455X (gfx1250) — compile-verified
//
#include <hip/hip_runtime.h>
#include <hip/hip_bf16.h>

#define N_NODES 100000
#define N_EDGES 1000000
#define FEA_DIM 64
#define H_DIM 128
#define LABEL_DIM 10
#define N_LAYER 3
#define N_SEL 50000
#define N_GRAPHS 64

typedef float v2f __attribute__((ext_vector_type(2)));
typedef float v8f __attribute__((ext_vector_type(8)));

// ---------------------------------------------------------------------------
// Degree / normalization kernels
// ---------------------------------------------------------------------------
__global__ void init_deg_kernel(float* __restrict__ deg, int n) {
    int i = blockIdx.x * blockDim.x + threadIdx.x;
    if (i < n) deg[i] = 1.0f;  // self-loop contributes 1
}

__global__ void count_deg_kernel(const int* __restrict__ dst,
                                 float* __restrict__ deg, int e) {
    int i = blockIdx.x * blockDim.x + threadIdx.x;
    if (i < e) atomicAdd(&deg[dst[i]], 1.0f);
}

__global__ void rsqrt_deg_kernel(float* __restrict__ deg, int n) {
    int i = blockIdx.x * blockDim.x + threadIdx.x;
    if (i < n) deg[i] = rsqrtf(deg[i]);
}

__global__ void zero_f32_kernel(float* __restrict__ p, long n) {
    long i = (long)blockIdx.x * blockDim.x + threadIdx.x;
    if (i < n) p[i] = 0.0f;
}

// ---------------------------------------------------------------------------
// GEMM: h[N,64] = x[N,64] @ W[64,64] using V_WMMA_F32_16X16X4_F32 (pure f32).
// One wave computes a 16x64 output strip: 4 accumulators, 16 K-steps of K=4.
// Layouts per CDNA5 ISA 7.12.2:
//   A vgpr v, lane l : m = l&15,        k = (l>>4)*2 + v
//   B vgpr v, lane l : k = (l>>4)*2+v,  n = l&15
//   D vgpr v, lane l : m = (l>>4)*8+v,  n = l&15
// ---------------------------------------------------------------------------
__global__ __launch_bounds__(256) void gemm_wmma_n64_kernel(
    const float* __restrict__ x, const float* __restrict__ W,
    float* __restrict__ h, int nTilesM) {
    int wave = blockIdx.x * (blockDim.x >> 5) + (threadIdx.x >> 5);
    if (wave >= nTilesM) return;
    int lane = threadIdx.x & 31;
    int r16  = lane & 15;          // A-row within tile / B,C column
    int koff = (lane >> 4) * 2;    // K sub-offset for A/B fragments
    long m0  = (long)wave * 16;

    v8f acc0 = {0.f,0.f,0.f,0.f,0.f,0.f,0.f,0.f};
    v8f acc1 = acc0, acc2 = acc0, acc3 = acc0;

    const float* xrow = x + (m0 + r16) * FEA_DIM;
#pragma unroll
    for (int kk = 0; kk < FEA_DIM; kk += 4) {
        v2f a;
        a.x = xrow[kk + koff + 0];
        a.y = xrow[kk + koff + 1];
        const float* wb = W + (kk + koff) * FEA_DIM + r16;
        v2f b0, b1, b2, b3;
        b0.x = wb[0];       b0.y = wb[FEA_DIM + 0];
        b1.x = wb[16];      b1.y = wb[FEA_DIM + 16];
        b2.x = wb[32];      b2.y = wb[FEA_DIM + 32];
        b3.x = wb[48];      b3.y = wb[FEA_DIM + 48];
        acc0 = __builtin_amdgcn_wmma_f32_16x16x4_f32(false, a, false, b0, (short)0, acc0, false, false);
        acc1 = __builtin_amdgcn_wmma_f32_16x16x4_f32(false, a, false, b1, (short)0, acc1, false, false);
        acc2 = __builtin_amdgcn_wmma_f32_16x16x4_f32(false, a, false, b2, (short)0, acc2, false, false);
        acc3 = __builtin_amdgcn_wmma_f32_16x16x4_f32(false, a, false, b3, (short)0, acc3, false, false);
    }

    int mrow = (lane >> 4) * 8;
    int ncol = lane & 15;
#pragma unroll
    for (int v = 0; v < 8; ++v) {
        float* hr = h + (m0 + mrow + v) * FEA_DIM + ncol;
        hr[0]  = acc0[v];
        hr[16] = acc1[v];
        hr[32] = acc2[v];
        hr[48] = acc3[v];
    }
}

// ---------------------------------------------------------------------------
// Edge scatter: xout[dst] += h[src] * dinv[src]*dinv[dst]  (plus self-loops).
// 16 lanes per edge, float4 per lane, f32 global atomics (L2-resident set).
// ---------------------------------------------------------------------------
__global__ void scatter_kernel(const float* __restrict__ h,
                               const float* __restrict__ dinv,
                               const int* __restrict__ esrc,
                               const int* __restrict__ edst,
                               float* __restrict__ xout,
                               int E, int N) {
    long t = (long)blockIdx.x * blockDim.x + threadIdx.x;
    long e = t >> 4;
    if (e >= (long)E + N) return;
    int j = (int)(t & 15) * 4;
    int s, d;
    if (e < E) { s = esrc[e]; d = edst[e]; }
    else       { s = d = (int)(e - E); }
    float c = dinv[s] * dinv[d];
    float4 hv = *(const float4*)(h + (long)s * FEA_DIM + j);
    float* o = xout + (long)d * FEA_DIM + j;
    atomicAdd(o + 0, hv.x * c);
    atomicAdd(o + 1, hv.y * c);
    atomicAdd(o + 2, hv.z * c);
    atomicAdd(o + 3, hv.w * c);
}

__global__ void bias_relu_kernel(float* __restrict__ x,
                                 const float* __restrict__ b, long total) {
    long i = (long)blockIdx.x * blockDim.x + threadIdx.x;
    if (i < total) {
        float v = x[i] + b[i & (FEA_DIM - 1)];
        x[i] = v > 0.0f ? v : 0.0f;
    }
}

// ---------------------------------------------------------------------------
// Pooling: atomic scatter of selected rows into pooled[64,64] + counts[64].
// ---------------------------------------------------------------------------
__global__ void pool_scatter_kernel(const float* __restrict__ x,
                                    const int* __restrict__ sel,
                                    const int* __restrict__ batch,
                                    float* __restrict__ pooled,
                                    float* __restrict__ cnt, int S) {
    long t = (long)blockIdx.x * blockDim.x + threadIdx.x;
    long s = t >> 4;
    if (s >= S) return;
    int j = (int)(t & 15) * 4;
    int n = sel[s];
    int g = batch[n];
    float4 v = *(const float4*)(x + (long)n * FEA_DIM + j);
    float* o = pooled + g * FEA_DIM + j;
    atomicAdd(o + 0, v.x);
    atomicAdd(o + 1, v.y);
    atomicAdd(o + 2, v.z);
    atomicAdd(o + 3, v.w);
    if ((t & 15) == 0) atomicAdd(&cnt[g], 1.0f);
}

// ---------------------------------------------------------------------------
// MLP head: per-graph mean -> lin1(64->128)+ReLU -> lin2(128->10). Tiny.
// ---------------------------------------------------------------------------
__global__ __launch_bounds__(128) void mlp_head_kernel(
    const float* __restrict__ pooled, const float* __restrict__ cnt,
    const float* __restrict__ W1, const float* __restrict__ b1,
    const float* __restrict__ W2, const float* __restrict__ b2,
    float* __restrict__ out) {
    __shared__ float pm[FEA_DIM];
    __shared__ float hid[H_DIM];
    int g = blockIdx.x;
    int tid = threadIdx.x;
    if (tid < FEA_DIM) {
        float c = cnt[g];
        c = c > 1.0f ? c : 1.0f;
        pm[tid] = pooled[g * FEA_DIM + tid] / c;
    }
    __syncthreads();
    float acc = b1[tid];
#pragma unroll
    for (int k = 0; k < FEA_DIM; ++k) acc += pm[k] * W1[k * H_DIM + tid];
    hid[tid] = acc > 0.0f ? acc : 0.0f;
    __syncthreads();
    if (tid < LABEL_DIM) {
        float o = b2[tid];
#pragma unroll
        for (int k = 0; k < H_DIM; ++k) o += hid[k] * W2[k * LABEL_DIM + tid];
        out[g * LABEL_DIM + tid] = o;
    }
}

// ---------------------------------------------------------------------------
extern "C" void kernel_launch(void* const* d_in, const int* in_sizes, int n_in,
                              void* d_out, int out_size, void* d_ws, size_t ws_size,
                              hipStream_t stream) {
    const float* x      = (const float*)d_in[0];
    const int*   eidx   = (const int*)  d_in[1];
    const int*   sel    = (const int*)  d_in[2];
    const int*   batch  = (const int*)  d_in[3];
    const float* convW  = (const float*)d_in[4];
    const float* convB  = (const float*)d_in[5];
    const float* W1     = (const float*)d_in[6];
    const float* b1     = (const float*)d_in[7];
    const float* W2     = (const float*)d_in[8];
    const float* b2     = (const float*)d_in[9];
    const int* esrc = eidx;              // edge_index[0]
    const int* edst = eidx + N_EDGES;    // edge_index[1]

    const long NF = (long)N_NODES * FEA_DIM;
    float* ws     = (float*)d_ws;
    float* bufH   = ws;                 // [N,64] GEMM output
    float* bufX1  = bufH  + NF;         // [N,64] ping
    float* bufX2  = bufX1 + NF;         // [N,64] pong
    float* dinv   = bufX2 + NF;         // [N]
    float* pooled = dinv + N_NODES;     // [64,64]
    float* cnt    = pooled + N_GRAPHS * FEA_DIM; // [64]

    const int TB = 256;
    // --- degree / dinv ---
    init_deg_kernel<<<(N_NODES + TB - 1) / TB, TB, 0, stream>>>(dinv, N_NODES);
    count_deg_kernel<<<(N_EDGES + TB - 1) / TB, TB, 0, stream>>>(edst, dinv, N_EDGES);
    rsqrt_deg_kernel<<<(N_NODES + TB - 1) / TB, TB, 0, stream>>>(dinv, N_NODES);

    // --- 3 GCN layers ---
    const int nTilesM = N_NODES / 16;                 // 6250
    const int gemmBlocks = (nTilesM + 7) / 8;         // 8 waves/block
    const long scatterThreads = ((long)N_EDGES + N_NODES) * 16;
    const long scatterBlocks = (scatterThreads + TB - 1) / TB;
    const long nfBlocks = (NF + TB - 1) / TB;

    const float* xin = x;
    for (int L = 0; L < N_LAYER; ++L) {
        float* xout = (L == 1) ? bufX2 : bufX1;
        gemm_wmma_n64_kernel<<<gemmBlocks, TB, 0, stream>>>(
            xin, convW + (long)L * FEA_DIM * FEA_DIM, bufH, nTilesM);
        zero_f32_kernel<<<nfBlocks, TB, 0, stream>>>(xout, NF);
        scatter_kernel<<<scatterBlocks, TB, 0, stream>>>(
            bufH, dinv, esrc, edst, xout, N_EDGES, N_NODES);
        bias_relu_kernel<<<nfBlocks, TB, 0, stream>>>(xout, convB + (long)L * FEA_DIM, NF);
        xin = xout;
    }

    // --- pooling ---
    zero_f32_kernel<<<(N_GRAPHS * FEA_DIM + N_GRAPHS + TB - 1) / TB, TB, 0, stream>>>(
        pooled, N_GRAPHS * FEA_DIM + N_GRAPHS);
    pool_scatter_kernel<<<((long)N_SEL * 16 + TB - 1) / TB, TB, 0, stream>>>(
        xin, sel, batch, pooled, cnt, N_SEL);

    // --- MLP head ---
    mlp_head_kernel<<<N_GRAPHS, H_DIM, 0, stream>>>(
        pooled, cnt, W1, b1, W2, b2, (float*)d_out);
}